// Leech2QuantizerUnitVol_59785944760427
// MI455X (gfx1250) — compile-verified
//
#include <hip/hip_runtime.h>

typedef __attribute__((ext_vector_type(2))) float v2f;
typedef __attribute__((ext_vector_type(8))) float v8f;

#define A_CONST 0.35355339059327373f   // np.float32(1/sqrt(8))

// Quantize (u - shift) to nearest D8 lattice point.
// Returns squared distance; if WANT_Y, yo = chosen point + shift.
template <bool WANT_Y>
__device__ __forceinline__ float d8_branch(const float u[8], float shift, float yo[8]) {
  float f[8], d[8];
  float sum = 0.f;
  float mx = -1.f;
  float dk = 0.f;
  int   k  = 0;
#pragma unroll
  for (int i = 0; i < 8; ++i) {
    float ui = u[i] - shift;
    float fi = rintf(ui);          // v_rndne_f32: round-half-even == jnp.round
    float di = ui - fi;
    f[i] = fi; d[i] = di;
    sum += fi;
    float ad = fabsf(di);
    if (ad > mx) { mx = ad; k = i; dk = di; }   // strict '>' keeps FIRST max (jnp.argmax)
  }
  float dist = 0.f;
#pragma unroll
  for (int i = 0; i < 8; ++i) dist += d[i] * d[i];
  const int   odd = ((int)sum) & 1;             // parity of integer sum (exact in f32 here)
  const float fix = (dk < 0.f) ? -1.f : 1.f;
  if (odd) {
    float nd = dk - fix;                        // replace d[k]^2 with (d[k]-fix)^2
    dist += nd * nd - dk * dk;
  }
  if (WANT_Y) {
#pragma unroll
    for (int i = 0; i < 8; ++i) {
      float yi = f[i] + shift;
      if (odd && (i == k)) yi += fix;           // predicated, no dynamic indexing
      yo[i] = yi;
    }
  }
  return dist;
}

// E8 quantize u (= v/4): nearer of D8 and D8+0.5 cosets. Returns sq dist in u-space.
template <bool WANT_Y>
__device__ __forceinline__ float e8_quant(const float u[8], float y[8]) {
  float y1[8], y2[8];
  float da = d8_branch<WANT_Y>(u, 0.0f, y1);
  float db = d8_branch<WANT_Y>(u, 0.5f, y2);
  bool p1 = (da <= db);                          // reference: d1 <= d2 -> coset 1
  if (WANT_Y) {
#pragma unroll
    for (int i = 0; i < 8; ++i) y[i] = p1 ? y1[i] : y2[i];
  }
  return p1 ? da : db;
}

__global__ __launch_bounds__(32)
void leech2_quant_kernel(const float* __restrict__ x_in,
                         const float* __restrict__ AT,
                         const int*   __restrict__ idx_c,
                         float*       __restrict__ out,
                         int B) {
  // Per-row tables, stored transposed: sD[m][t*16 + i] = d_m[row i*16+t]
  __shared__ float sD[3][256];
  __shared__ int   sTab[256];

  const int b = blockIdx.x;
  if (b >= B) return;
  const int lane = threadIdx.x;   // wave32: one wave per row

  // Stage TABLE[pi,pj] from idx_c (idx_c[(pi*16+pj)*16 + t] is t-invariant).
#pragma unroll
  for (int j = 0; j < 8; ++j) {
    int i = lane * 8 + j;
    sTab[i] = idx_c[i << 4];
  }

  // x = x_in / a (all lanes load the same 24 values: L0 broadcast).
  float x[24];
#pragma unroll
  for (int i = 0; i < 24; ++i) x[i] = x_in[b * 24 + i] / A_CONST;

  // ---- Stage 1: build the three 256-entry distance tables (8 rows/lane) ----
#pragma unroll
  for (int j = 0; j < 8; ++j) {
    const int r = lane + (j << 5);
    float at[8];
#pragma unroll
    for (int i = 0; i < 8; ++i) at[i] = AT[r * 8 + i];
    const int t  = r & 15;
    const int pi = r >> 4;
#pragma unroll
    for (int m = 0; m < 3; ++m) {
      float u[8];
#pragma unroll
      for (int i = 0; i < 8; ++i) u[i] = (x[m * 8 + i] - at[i]) * 0.25f;
      // store sq-dist in u-space (uniform 16x factor cancels in argmin)
      sD[m][t * 16 + pi] = e8_quant<false>(u, nullptr);
    }
  }
  __syncthreads();

  // ---- Stage 2: 4096-candidate search via WMMA outer-sums ----
  // Per t: S(pi,pj) = d1[pi,t] + d2[pj,t] as A(16x4) x B(4x16) rank-2 product.
  // Argmin via packed u64 key: dist >= 0 so IEEE bits are order-monotonic;
  // key = (dist_bits << 32) | cand reproduces (dist, first-index) ordering exactly.
  unsigned long long bestk = ~0ULL;
  const int  pjn = lane & 15;
  const int  hi8 = (lane >> 4) << 3;   // D layout: M = v + 8*(lane>=16)
  const bool lo  = lane < 16;
  const unsigned lanebase = (unsigned)((hi8 << 8) | (pjn << 4));

  for (int t = 0; t < 16; ++t) {
    const float d1v = sD[0][t * 16 + pjn];
    const float d2v = sD[1][t * 16 + pjn];
    // A VGPR0: lanes0-15 K=0 (d1 col), lanes16-31 K=2 (0); A VGPR1: K=1 (ones), K=3 (0)
    v2f Am, Bm;
    Am[0] = lo ? d1v : 0.f;
    Am[1] = lo ? 1.f : 0.f;
    // B VGPR0: lanes0-15 K=0 row (ones); B VGPR1: lanes0-15 K=1 row (d2)
    Bm[0] = lo ? 1.f : 0.f;
    Bm[1] = lo ? d2v : 0.f;
    v8f c = {};
    c = __builtin_amdgcn_wmma_f32_16x16x4_f32(
        /*neg_a=*/false, Am, /*neg_b=*/false, Bm,
        /*c_mod=*/(short)0, c, /*reuse_a=*/false, /*reuse_b=*/false);
#pragma unroll
    for (int v = 0; v < 8; ++v) {
      const int   pi  = v + hi8;
      const int   pq  = sTab[pi * 16 + pjn];
      const float tot = c[v] + sD[2][t * 16 + pq];
      // cand = ((pi<<4 | pj) << 4) | t  (reference candidate order), bit-disjoint sum:
      const unsigned cand = (unsigned)(v << 8) + lanebase + (unsigned)t;
      const unsigned long long key =
          ((unsigned long long)__float_as_uint(tot) << 32) | cand;
      bestk = (key < bestk) ? key : bestk;   // v_cmp_lt_u64 + 2 cndmask
    }
  }

  // ---- wave32 min-reduction (u64 min preserves dist-then-index order) ----
#pragma unroll
  for (int off = 16; off > 0; off >>= 1) {
    unsigned long long ok = (unsigned long long)
        __shfl_xor((long long)bestk, off, 32);
    bestk = (ok < bestk) ? ok : bestk;
  }
  const int bestc = (int)(bestk & 0xFFFFu);

  // ---- winner: lanes 0..2 re-quantize their 8-block and emit ----
  if (lane < 3) {
    const int t   = bestc & 15;
    const int pjw = (bestc >> 4) & 15;
    const int piw = (bestc >> 8) & 15;
    int r;
    if (lane == 0)      r = (piw << 4) | t;
    else if (lane == 1) r = (pjw << 4) | t;
    else                r = (sTab[(piw << 4) | pjw] << 4) | t;
    float u[8], y[8];
#pragma unroll
    for (int i = 0; i < 8; ++i) u[i] = (x[lane * 8 + i] - AT[r * 8 + i]) * 0.25f;
    e8_quant<true>(u, y);
#pragma unroll
    for (int i = 0; i < 8; ++i)
      out[b * 24 + lane * 8 + i] = (4.f * y[i] + AT[r * 8 + i]) * A_CONST;
  }
}

extern "C" void kernel_launch(void* const* d_in, const int* in_sizes, int n_in,
                              void* d_out, int out_size, void* d_ws, size_t ws_size,
                              hipStream_t stream) {
  const float* x_in  = (const float*)d_in[0];   // (B, 24) f32
  const float* AT    = (const float*)d_in[1];   // (256, 8) f32
  const int*   idx_c = (const int*)d_in[4];     // (4096,) i32
  float*       out   = (float*)d_out;           // (B, 24) f32
  const int B = in_sizes[0] / 24;
  leech2_quant_kernel<<<B, 32, 0, stream>>>(x_in, AT, idx_c, out, B);
}